// PlaceGCN_49168785605218
// MI455X (gfx1250) — compile-verified
//
#include <hip/hip_runtime.h>

typedef __attribute__((ext_vector_type(2))) float v2f;
typedef __attribute__((ext_vector_type(8))) float v8f;

// ---------------------------------------------------------------------------
// Zero-fill scratch accumulators (segment_sum needs zeroed destinations).
// ---------------------------------------------------------------------------
__global__ void zero_f32(float* __restrict__ p, long n) {
    long i = (long)blockIdx.x * blockDim.x + threadIdx.x;
    if (i < n) p[i] = 0.0f;
}

// ---------------------------------------------------------------------------
// Edge-parallel GCN aggregation: out[dst[e], :] += feat[src[e], :]
// One wave (32 lanes) per edge; lane l handles V consecutive floats
// (V=4 -> float4 load + 4 f32 atomics; V=2 -> float2 load + 2 f32 atomics).
// All traffic stays in L2 (feat + out accumulators << 192 MB).
// ---------------------------------------------------------------------------
template <int V>  // floats per lane; feature width F = 32*V
__global__ void gcn_aggregate(const float* __restrict__ feat,
                              const int* __restrict__ src,
                              const int* __restrict__ dst,
                              float* __restrict__ out,
                              int n_edges) {
    constexpr int F = 32 * V;
    long tid  = (long)blockIdx.x * blockDim.x + threadIdx.x;
    int  lane = (int)(tid & 31);
    long edge = tid >> 5;
    if (edge >= n_edges) return;               // wave-uniform exit

    // edge index is wave-uniform -> force scalar regs for the node ids
    int s = __builtin_amdgcn_readfirstlane(src[edge]);
    int d = __builtin_amdgcn_readfirstlane(dst[edge]);

    const float* fp = feat + (long)s * F + lane * V;
    float*       op = out  + (long)d * F + lane * V;

    float v[V];
    if constexpr (V == 4) {
        float4 t = *(const float4*)fp;
        v[0] = t.x; v[1] = t.y; v[2] = t.z; v[3] = t.w;
    } else if constexpr (V == 2) {
        float2 t = *(const float2*)fp;
        v[0] = t.x; v[1] = t.y;
    } else {
#pragma unroll
        for (int i = 0; i < V; ++i) v[i] = fp[i];
    }
#pragma unroll
    for (int i = 0; i < V; ++i) atomicAdd(&op[i], v[i]);
}

// ---------------------------------------------------------------------------
// D = A[M,K] x B[K,N] + bias (optional ReLU), f32 end-to-end via
// V_WMMA_F32_16X16X4_F32. One wave computes one 16x16 output tile.
//
// Per ISA 7.12.2 (32-bit layouts, wave32):
//   A 16x4 :  lanes 0-15 -> M=lane,   {v0,v1} = K0,K1
//             lanes16-31 -> M=lane-16,{v0,v1} = K2,K3
//   B 4x16 :  lanes 0-15 -> N=lane,   {v0,v1} = K0,K1 rows
//             lanes16-31 -> N=lane-16,{v0,v1} = K2,K3 rows
//   C/D    :  vgpr r, lanes 0-15 -> (M=r,   N=lane)
//                     lanes16-31 -> (M=r+8, N=lane-16)
// ---------------------------------------------------------------------------
__global__ void gemm_bias_wmma(const float* __restrict__ A,   // [M x K] row-major
                               const float* __restrict__ B,   // [K x N] row-major
                               const float* __restrict__ bias,// [N]
                               float* __restrict__ D,         // [M x N]
                               int M, int K, int N, int do_relu) {
    int wave = (int)((blockIdx.x * (long)blockDim.x + threadIdx.x) >> 5);
    int lane = threadIdx.x & 31;
    int tiles_n = N >> 4;
    int tiles_m = M >> 4;
    if (wave >= tiles_m * tiles_n) return;     // wave-uniform exit, EXEC stays full

    int tm = wave / tiles_n;
    int tn = wave - tm * tiles_n;
    int m0 = tm << 4, n0 = tn << 4;

    int half = lane >> 4;     // 0: K{0,1}, 1: K{2,3} of each K-step
    int l    = lane & 15;
    int arow = m0 + l;        // A row owned by this lane
    int bcol = n0 + l;        // B/D column owned by this lane

    const float* Arow = A + (size_t)arow * K;

    v8f acc = {};
    for (int k0 = 0; k0 < K; k0 += 4) {
        int ka = k0 + half * 2;
        v2f a, b;
        a.x = Arow[ka];
        a.y = Arow[ka + 1];
        b.x = B[(size_t)ka * N + bcol];
        b.y = B[(size_t)(ka + 1) * N + bcol];
        // 8 args: (neg_a, A, neg_b, B, c_mod, C, reuse_a, reuse_b)
        acc = __builtin_amdgcn_wmma_f32_16x16x4_f32(
            /*neg_a=*/false, a, /*neg_b=*/false, b,
            /*c_mod=*/(short)0, acc, /*reuse_a=*/false, /*reuse_b=*/false);
    }

    float bv = bias[bcol];
    int rbase = m0 + half * 8;
#pragma unroll
    for (int r = 0; r < 8; ++r) {
        float v = acc[r] + bv;
        if (do_relu) v = fmaxf(v, 0.0f);
        D[(size_t)(rbase + r) * N + bcol] = v;
    }
}

// ---------------------------------------------------------------------------
// Host-side orchestration (all launches on `stream`, graph-capture safe).
// ---------------------------------------------------------------------------
extern "C" void kernel_launch(void* const* d_in, const int* in_sizes, int n_in,
                              void* d_out, int out_size, void* d_ws, size_t ws_size,
                              hipStream_t stream) {
    const float* features = (const float*)d_in[0];
    const float* W1       = (const float*)d_in[1];
    const float* b1       = (const float*)d_in[2];
    const float* W2       = (const float*)d_in[3];
    const float* b2       = (const float*)d_in[4];
    const int*   src      = (const int*)d_in[5];
    const int*   dst      = (const int*)d_in[6];
    float*       out      = (float*)d_out;

    const int H1 = in_sizes[2];            // 64
    const int H2 = in_sizes[4];            // 32
    const int F0 = in_sizes[1] / H1;       // 128
    const int N  = in_sizes[0] / F0;       // 20000
    const int E  = in_sizes[5];            // 640000

    float* h1 = (float*)d_ws;              // [N x F0] aggregated input feats
    float* x  = h1 + (size_t)N * F0;       // [N x H1] layer-1 activations
    float* h2 = x  + (size_t)N * H1;       // [N x H1] aggregated activations

    // Zero the two segment-sum accumulators.
    {
        long n1 = (long)N * F0;
        zero_f32<<<(int)((n1 + 255) / 256), 256, 0, stream>>>(h1, n1);
        long n2 = (long)N * H1;
        zero_f32<<<(int)((n2 + 255) / 256), 256, 0, stream>>>(h2, n2);
    }

    // Layer 1 aggregation: h1[dst] += features[src]  (128 feats: V=4)
    {
        long threads = (long)E * 32;
        gcn_aggregate<4><<<(int)((threads + 255) / 256), 256, 0, stream>>>(
            features, src, dst, h1, E);
    }

    // Layer 1 dense: x = relu(h1 @ W1 + b1) via f32 WMMA
    {
        long waves = (long)(N / 16) * (H1 / 16);
        long threads = waves * 32;
        gemm_bias_wmma<<<(int)((threads + 255) / 256), 256, 0, stream>>>(
            h1, W1, b1, x, N, F0, H1, /*relu=*/1);
    }

    // Layer 2 aggregation: h2[dst] += x[src]  (64 feats: V=2)
    {
        long threads = (long)E * 32;
        gcn_aggregate<2><<<(int)((threads + 255) / 256), 256, 0, stream>>>(
            x, src, dst, h2, E);
    }

    // Layer 2 dense: out = h2 @ W2 + b2 via f32 WMMA
    {
        long waves = (long)(N / 16) * (H2 / 16);
        long threads = waves * 32;
        gemm_bias_wmma<<<(int)((threads + 255) / 256), 256, 0, stream>>>(
            h2, W2, b2, out, N, H1, H2, /*relu=*/0);
    }
}